// TSGCNet_28853590295300
// MI455X (gfx1250) — compile-verified
//
#include <hip/hip_runtime.h>

// ---------------- constants ----------------
#define NPTS   8000          // points per batch
#define NB     2             // batch
#define KNBR   16            // k neighbors
#define NC_    2000          // N-chunk for graph pipeline (8000/2000 = 4 chunks)
#define NCK_   (NC_*KNBR)    // 32000 columns per chunked GEMM (multiple of 16)

#define ACT_NONE  0
#define ACT_RELU  1
#define ACT_LEAKY 2

typedef float v2f __attribute__((ext_vector_type(2)));
typedef float v8f __attribute__((ext_vector_type(8)));

// ---------------------------------------------------------------------------
// WMMA f32 GEMM with M-register-blocking (MT tiles of 16 rows per wave):
//   out[row+rowOff, n] = act(S[row]*(W·X)[row,n] + Bb[row]) (* G)
// Caller guarantees: O % (16*MT) == 0, I % 4 == 0, cols % 16 == 0.
// Per K-step: 1 B-frag (2 x b32) reused by MT WMMAs, MT A-frags (b64 each)
//   -> 1.5 VMEM per v_wmma at MT=4 vs 3.0 at MT=1 (issue-rate limited kernel).
// MT independent accumulators avoid back-to-back WMMA RAW on the same D.
// A-frag (16x4 f32): lane<16 -> M=lane,K={0,1}; lane>=16 -> K={2,3}.
// B-frag (4x16 f32): mirrored. C/D: 8 VGPRs, M = r + 8*(lane>>4).
// ---------------------------------------------------------------------------
template <int MT>
__global__ void gemm16_wmma(const float* __restrict__ W, const float* __restrict__ X,
                            const float* __restrict__ S, const float* __restrict__ Bb,
                            const float* __restrict__ G, float* __restrict__ out,
                            int O, int I, int cols, int ldo, int rowOff, int act)
{
  const int o0   = blockIdx.x * (16 * MT);
  const int n0   = blockIdx.y * 16;
  const int lane = threadIdx.x & 31;
  const int half = lane >> 4;
  const int l15  = lane & 15;
  const int n    = n0 + l15;
  const int koff = half * 2;

  v8f acc[MT];
#pragma unroll
  for (int mt = 0; mt < MT; ++mt) acc[mt] = (v8f){};

  for (int kk = 0; kk < I; kk += 4) {
    v2f b;
    b.x = X[(kk + koff    ) * cols + n];
    b.y = X[(kk + koff + 1) * cols + n];
    if (kk + 4 < I) {
      // next K-step's B rows: streaming operand -> hint into cache early
      __builtin_prefetch(X + (kk + 4 + koff) * cols + n, 0, 3);
      __builtin_prefetch(X + (kk + 5 + koff) * cols + n, 0, 3);
    }
#pragma unroll
    for (int mt = 0; mt < MT; ++mt) {
      v2f a = *(const v2f*)(W + (o0 + mt * 16 + l15) * I + kk + koff); // 8B aligned
      acc[mt] = __builtin_amdgcn_wmma_f32_16x16x4_f32(false, a, false, b,
                                                      (short)0, acc[mt], false, false);
    }
  }
#pragma unroll
  for (int mt = 0; mt < MT; ++mt) {
#pragma unroll
    for (int r = 0; r < 8; ++r) {
      int   row = o0 + mt * 16 + r + 8 * half;
      float v   = acc[mt][r];
      if (S)                 v = v * S[row] + Bb[row];
      if (act == ACT_RELU)   v = fmaxf(v, 0.0f);
      else if (act == ACT_LEAKY) v = (v > 0.0f) ? v : 0.2f * v;
      if (G)                 v *= G[row * ldo + n];
      out[(row + rowOff) * ldo + n] = v;
    }
  }
}

// ---------------------------------------------------------------------------
// Fused conv(128->1024)+BN+ReLU+max over N (STN global feature).
// One block per output channel; avoids a 32MB/batch HBM round trip.
// ---------------------------------------------------------------------------
__global__ void gemm_colmax(const float* __restrict__ W, const float* __restrict__ X,
                            const float* __restrict__ S, const float* __restrict__ Bb,
                            float* __restrict__ out, int I, int N)
{
  const int c = blockIdx.x;
  const int t = threadIdx.x;
  float best = -3.4e38f;
  for (int n = t; n < N; n += 256) {
    float v = 0.0f;
    for (int i = 0; i < I; ++i) v += W[c * I + i] * X[i * N + n];
    v = fmaxf(v * S[c] + Bb[c], 0.0f);
    best = fmaxf(best, v);
  }
  __shared__ float red[256];
  red[t] = best;
  __syncthreads();
  for (int s = 128; s > 0; s >>= 1) {
    if (t < s) red[t] = fmaxf(red[t], red[t + s]);
    __syncthreads();
  }
  if (t == 0) out[c] = red[0];
}

// Small FC: out[o] = act(W[o,:]·in + bias[o]) (+ eye(12) flattened if addEye)
__global__ void fc_kernel(const float* __restrict__ W, const float* __restrict__ bias,
                          const float* __restrict__ in, float* __restrict__ out,
                          int O, int I, int act, int addEye)
{
  int o = blockIdx.x * blockDim.x + threadIdx.x;
  if (o >= O) return;
  float v = bias[o];
  for (int i = 0; i < I; ++i) v += W[o * I + i] * in[i];
  if (act) v = fmaxf(v, 0.0f);
  if (addEye && (o % 13 == 0)) v += 1.0f;   // eye(12).reshape(-1)
  out[o] = v;
}

// coor'[d,n] = sum_c coor[c,n] * T[c*12+d]
__global__ void apply_transform(const float* __restrict__ x, const float* __restrict__ T,
                                float* __restrict__ out, int N)
{
  int n = blockIdx.x * blockDim.x + threadIdx.x;
  if (n >= N) return;
  float p[12];
#pragma unroll
  for (int c = 0; c < 12; ++c) p[c] = x[c * N + n];
#pragma unroll
  for (int d = 0; d < 12; ++d) {
    float v = 0.0f;
#pragma unroll
    for (int c = 0; c < 12; ++c) v += p[c] * T[c * 12 + d];
    out[d * N + n] = v;
  }
}

// ---------------------------------------------------------------------------
// kNN: per-point top-(k+1) of pd = -||p_n - p_m||^2, drop self (slot 0).
// pts (C x N) is <=2MB -> L2 resident; pts[c*N+m] is wave-uniform (scalar$).
// ---------------------------------------------------------------------------
__global__ void knn_topk(const float* __restrict__ pts, int C, int N,
                         int* __restrict__ idx)
{
  int n = blockIdx.x * blockDim.x + threadIdx.x;
  if (n >= N) return;
  float pn[64];
  for (int c = 0; c < C; ++c) pn[c] = pts[c * N + n];
  float bd[KNBR + 1];
  int   bi[KNBR + 1];
  for (int j = 0; j <= KNBR; ++j) { bd[j] = -3.4e38f; bi[j] = 0; }
  for (int m = 0; m < N; ++m) {
    float d = 0.0f;
    for (int c = 0; c < C; ++c) { float t = pn[c] - pts[c * N + m]; d -= t * t; }
    if (d > bd[KNBR]) {                 // stable descending insertion (ties keep lower m)
      int j = KNBR;
      while (j > 0 && bd[j - 1] < d) { bd[j] = bd[j - 1]; bi[j] = bi[j - 1]; --j; }
      bd[j] = d; bi[j] = m;
    }
  }
  for (int j = 0; j < KNBR; ++j) idx[n * KNBR + j] = bi[j + 1];
}

// cf[c, nn*K + j] = (c<C) ? pts[c, idx[n,j]] : pts[c-C, n]   (n = n0+nn)
__global__ void build_cf(const float* __restrict__ pts, const int* __restrict__ idx,
                         int C, int n0, float* __restrict__ out)
{
  int t = blockIdx.x * blockDim.x + threadIdx.x;
  int total = 2 * C * NCK_;
  if (t >= total) return;
  int c   = t / NCK_;
  int rem = t % NCK_;
  int nn  = rem / KNBR;
  int j   = rem % KNBR;
  int n   = n0 + nn;
  float v = (c < C) ? pts[c * NPTS + idx[n * KNBR + j]]
                    : pts[(c - C) * NPTS + n];
  out[t] = v;
}

// delta[c, j*NC + nn] = (c<C) ? pts[c,n]-pts[c,nbr] : pts[c-C,nbr]
__global__ void build_delta(const float* __restrict__ pts, const int* __restrict__ idx,
                            int C, int n0, float* __restrict__ out)
{
  int t = blockIdx.x * blockDim.x + threadIdx.x;
  int total = 2 * C * KNBR * NC_;
  if (t >= total) return;
  int c   = t / (KNBR * NC_);
  int rem = t % (KNBR * NC_);
  int j   = rem / NC_;
  int nn  = rem % NC_;
  int n   = n0 + nn;
  int nb  = idx[n * KNBR + j];
  float v = (c < C) ? (pts[c * NPTS + n] - pts[c * NPTS + nb])
                    : pts[(c - C) * NPTS + nb];
  out[t] = v;
}

// out[rowOff+c, n0+nn] = sum_j softmax_j(e[c, j*NC+nn]) * feat[c, nn*K+j]
__global__ void att_combine(const float* __restrict__ e, const float* __restrict__ feat,
                            float* __restrict__ out, int Cp, int n0, int rowOff)
{
  int t = blockIdx.x * blockDim.x + threadIdx.x;
  if (t >= Cp * NC_) return;
  int c  = t / NC_;
  int nn = t % NC_;
  const float* ec = e + (size_t)c * KNBR * NC_ + nn;
  float mx = -3.4e38f;
#pragma unroll
  for (int j = 0; j < KNBR; ++j) mx = fmaxf(mx, ec[j * NC_]);
  float w[KNBR], s = 0.0f;
#pragma unroll
  for (int j = 0; j < KNBR; ++j) { w[j] = expf(ec[j * NC_] - mx); s += w[j]; }
  const float* fc = feat + (size_t)c * NC_ * KNBR + (size_t)nn * KNBR;
  float acc = 0.0f;
#pragma unroll
  for (int j = 0; j < KNBR; ++j) acc += (w[j] / s) * fc[j];
  out[(rowOff + c) * NPTS + n0 + nn] = acc;
}

// out[rowOff+c, n0+nn] = max_j feat[c, nn*K+j]
__global__ void max_k(const float* __restrict__ feat, float* __restrict__ out,
                      int Cp, int n0, int rowOff)
{
  int t = blockIdx.x * blockDim.x + threadIdx.x;
  if (t >= Cp * NC_) return;
  int c  = t / NC_;
  int nn = t % NC_;
  const float* fc = feat + (size_t)c * NC_ * KNBR + (size_t)nn * KNBR;
  float mx = -3.4e38f;
#pragma unroll
  for (int j = 0; j < KNBR; ++j) mx = fmaxf(mx, fc[j]);
  out[(rowOff + c) * NPTS + n0 + nn] = mx;
}

// final 128->2 projection + log_softmax, transposed write (B,N,2)
__global__ void pred4_lsm(const float* __restrict__ W, const float* __restrict__ b,
                          const float* __restrict__ h, float* __restrict__ out, int N)
{
  int n = blockIdx.x * blockDim.x + threadIdx.x;
  if (n >= N) return;
  float s0 = b[0], s1 = b[1];
  for (int i = 0; i < 128; ++i) {
    float x = h[i * N + n];
    s0 += W[i] * x;
    s1 += W[128 + i] * x;
  }
  float m = fmaxf(s0, s1);
  float l = m + logf(expf(s0 - m) + expf(s1 - m));
  out[n * 2 + 0] = s0 - l;
  out[n * 2 + 1] = s1 - l;
}

// ---------------------------------------------------------------------------
// Host orchestration
// ---------------------------------------------------------------------------
struct LayerP { const float *w, *s, *b; };
struct StnP   { const float *w1,*w2,*w3,*fc1w,*fc1b,*fc2w,*fc2b,*fc3w,*fc3b,
                            *s1,*b1,*s2,*b2,*s3,*b3; };

extern "C" void kernel_launch(void* const* d_in, const int* in_sizes, int n_in,
                              void* d_out, int out_size, void* d_ws, size_t ws_size,
                              hipStream_t stream)
{
  (void)in_sizes; (void)out_size; (void)ws_size;
  if (n_in < 78) return;
  const int N = NPTS;

  auto F = [&](int i) { return (const float*)d_in[i]; };

  // input order: depth-first insertion order of setup_inputs() dict
  const float* xin = F(0);
  auto loadStn = [&](int base) {
    StnP s;
    s.w1 = F(base + 0);  s.w2 = F(base + 1);  s.w3 = F(base + 2);
    s.fc1w = F(base + 3); s.fc1b = F(base + 4);
    s.fc2w = F(base + 5); s.fc2b = F(base + 6);
    s.fc3w = F(base + 7); s.fc3b = F(base + 8);
    s.s1 = F(base + 9);  s.b1 = F(base + 10);
    s.s2 = F(base + 11); s.b2 = F(base + 12);
    s.s3 = F(base + 13); s.b3 = F(base + 14);
    return s;
  };
  StnP stn_c = loadStn(1), stn_n = loadStn(16);
  auto loadL = [&](int base) { LayerP l; l.w = F(base); l.s = F(base+1); l.b = F(base+2); return l; };
  LayerP conv1c = loadL(31), conv2c = loadL(34), conv3c = loadL(37), conv4c = loadL(40);
  LayerP att1   = loadL(43), att2   = loadL(46), att3   = loadL(49);
  LayerP conv1n = loadL(52), conv2n = loadL(55), conv3n = loadL(58), conv4n = loadL(61);
  LayerP fa     = loadL(64), pred1  = loadL(67), pred2  = loadL(70), pred3  = loadL(73);
  const float* p4w = F(76); const float* p4b = F(77);

  // workspace carve-up (peak ~125MB, chunk pipeline sized for L2 residency)
  char* ws = (char*)d_ws; size_t off = 0;
  auto A = [&](size_t nfl) {
    float* p = (float*)(ws + off);
    off += ((nfl * 4 + 255) / 256) * 256;
    return p;
  };
  float* coorT = A((size_t)NB * 12 * N);
  float* norT  = A((size_t)NB * 12 * N);
  float* h1    = A((size_t)64  * N);          // per-batch reused
  float* h2    = A((size_t)128 * N);
  float* gbuf  = A(1024);
  float* f1    = A(512);
  float* f2    = A(256);
  float* Tm    = A(144);
  int*   idxb  = (int*)A((size_t)N * KNBR);   // per-batch reused
  float* ccin  = A((size_t)NB * 224 * N);     // rows: c1[0:32) c2[32:96) c3[96:224)
  float* nnin  = A((size_t)NB * 224 * N);
  float* hbuf  = A((size_t)NB * 512 * N);     // concat(conv4_c, conv4_n)
  float* bufA  = A((size_t)128 * NCK_);       // 4.096M floats each (== 512*8000)
  float* bufB  = A((size_t)128 * NCK_);
  float* bufC  = A((size_t)128 * NCK_);

  auto gemm = [&](const float* Wm, const float* Xm, const float* Sv, const float* Bv,
                  const float* Gv, float* Om, int O, int I, int cols,
                  int ldo, int rowOff, int act) {
    dim3 blk(32, 1, 1);
    if (O % 64 == 0)
      gemm16_wmma<4><<<dim3(O / 64, cols / 16, 1), blk, 0, stream>>>(
          Wm, Xm, Sv, Bv, Gv, Om, O, I, cols, ldo, rowOff, act);
    else if (O % 32 == 0)
      gemm16_wmma<2><<<dim3(O / 32, cols / 16, 1), blk, 0, stream>>>(
          Wm, Xm, Sv, Bv, Gv, Om, O, I, cols, ldo, rowOff, act);
    else
      gemm16_wmma<1><<<dim3(O / 16, cols / 16, 1), blk, 0, stream>>>(
          Wm, Xm, Sv, Bv, Gv, Om, O, I, cols, ldo, rowOff, act);
  };

  // ---- STN + transform for coor / nor -----------------------------------
  for (int b = 0; b < NB; ++b) {
    for (int half = 0; half < 2; ++half) {
      const StnP& sp = half ? stn_n : stn_c;
      const float* plane = xin + (size_t)b * 24 * N + (size_t)half * 12 * N;
      float* outP = (half ? norT : coorT) + (size_t)b * 12 * N;
      gemm(sp.w1, plane, sp.s1, sp.b1, nullptr, h1, 64, 12, N, N, 0, ACT_RELU);
      gemm(sp.w2, h1,    sp.s2, sp.b2, nullptr, h2, 128, 64, N, N, 0, ACT_RELU);
      gemm_colmax<<<1024, 256, 0, stream>>>(sp.w3, h2, sp.s3, sp.b3, gbuf, 128, N);
      fc_kernel<<<8, 64, 0, stream>>>(sp.fc1w, sp.fc1b, gbuf, f1, 512, 1024, 1, 0);
      fc_kernel<<<4, 64, 0, stream>>>(sp.fc2w, sp.fc2b, f1, f2, 256, 512, 1, 0);
      fc_kernel<<<3, 64, 0, stream>>>(sp.fc3w, sp.fc3b, f2, Tm, 144, 256, 0, 1);
      apply_transform<<<(N + 255) / 256, 256, 0, stream>>>(plane, Tm, outP, N);
    }
  }

  // ---- one graph round ---------------------------------------------------
  auto round = [&](const float* ptsC, size_t strC, const float* ptsN, size_t strN,
                   int C, LayerP cv, LayerP at, LayerP cn, int Cout, int rowOff) {
    for (int b = 0; b < NB; ++b) {
      const float* pc = ptsC + b * strC;
      const float* pn = ptsN + b * strN;
      knn_topk<<<(N + 127) / 128, 128, 0, stream>>>(pc, C, N, idxb);
      for (int ch = 0; ch < N / NC_; ++ch) {
        int n0 = ch * NC_;
        int tot = 2 * C * NCK_;
        build_cf<<<(tot + 255) / 256, 256, 0, stream>>>(pc, idxb, C, n0, bufA);
        gemm(cv.w, bufA, cv.s, cv.b, nullptr, bufB, Cout, 2 * C, NCK_, NCK_, 0, ACT_LEAKY);
        build_delta<<<(tot + 255) / 256, 256, 0, stream>>>(pc, idxb, C, n0, bufA);
        gemm(at.w, bufA, at.s, at.b, nullptr, bufC, Cout, 2 * C, NCK_, NCK_, 0, ACT_LEAKY);
        att_combine<<<(Cout * NC_ + 255) / 256, 256, 0, stream>>>(
            bufC, bufB, ccin + (size_t)b * 224 * N, Cout, n0, rowOff);
        build_cf<<<(tot + 255) / 256, 256, 0, stream>>>(pn, idxb, C, n0, bufA);
        gemm(cn.w, bufA, cn.s, cn.b, nullptr, bufB, Cout, 2 * C, NCK_, NCK_, 0, ACT_LEAKY);
        max_k<<<(Cout * NC_ + 255) / 256, 256, 0, stream>>>(
            bufB, nnin + (size_t)b * 224 * N, Cout, n0, rowOff);
      }
    }
  };

  size_t strCc = (size_t)224 * N;
  round(coorT, (size_t)12 * N, norT, (size_t)12 * N, 12, conv1c, att1, conv1n, 32, 0);
  round(ccin,            strCc, nnin,            strCc, 32, conv2c, att2, conv2n, 64, 32);
  round(ccin + 32 * N,   strCc, nnin + 32 * N,   strCc, 64, conv3c, att3, conv3n, 128, 96);

  // ---- head: conv4 / fa gate / pred chain / log-softmax ------------------
  for (int b = 0; b < NB; ++b) {
    float* hb = hbuf + (size_t)b * 512 * N;
    gemm(conv4c.w, ccin + (size_t)b * 224 * N, conv4c.s, conv4c.b, nullptr, hb, 256, 224, N, N, 0,   ACT_LEAKY);
    gemm(conv4n.w, nnin + (size_t)b * 224 * N, conv4n.s, conv4n.b, nullptr, hb, 256, 224, N, N, 256, ACT_LEAKY);
    gemm(fa.w,    hb,   fa.s,    fa.b,    hb,      bufA, 512, 512, N, N, 0, ACT_LEAKY); // cbl(h)*h
    gemm(pred1.w, bufA, pred1.s, pred1.b, nullptr, bufB, 512, 512, N, N, 0, ACT_LEAKY);
    gemm(pred2.w, bufB, pred2.s, pred2.b, nullptr, bufA, 256, 512, N, N, 0, ACT_LEAKY);
    gemm(pred3.w, bufA, pred3.s, pred3.b, nullptr, bufB, 128, 256, N, N, 0, ACT_LEAKY);
    pred4_lsm<<<(N + 255) / 256, 256, 0, stream>>>(
        p4w, p4b, bufB, (float*)d_out + (size_t)b * N * 2, N);
  }
}